// ChamferLossWholeImage_53206054863124
// MI455X (gfx1250) — compile-verified
//
#include <hip/hip_runtime.h>
#include <hip/hip_bf16.h>
#include <math.h>

typedef __attribute__((ext_vector_type(2))) float v2f;
typedef __attribute__((ext_vector_type(8))) float v8f;

#define W_IMG 256
#define BIG 3.0e38f

// ---------------------------------------------------------------------------
// init: out[0] = 0, rowmin[i] = +inf (uint bit pattern; d^2 clamped >= 0 so
// uint ordering == float ordering)
// ---------------------------------------------------------------------------
__global__ void chamfer_init(float* out, unsigned int* rowmin, int n) {
    int i = blockIdx.x * blockDim.x + threadIdx.x;
    if (i == 0) out[0] = 0.0f;
    if (i < n) rowmin[i] = 0x7F800000u;  // +inf
}

// ---------------------------------------------------------------------------
// Kernel A: per-pixel min over all points, fused sqrt+sum.
// One wave owns one 16-pixel tile (B fragment fixed), loops over 125 point
// tiles. d^2 tile = A(16x4) x B(4x16) per V_WMMA_F32_16X16X4_F32 using
//   A row (point M): [px, py, |p|^2, 1]
//   B col (pixel N): [-2cx, -2cy, 1, |c|^2]
// LDS holds the *lane-ready* A fragment: sfrag[0][i]={px,py} (lanes 0-15,
// K=0,1), sfrag[1][i]={pp,1} (lanes 16-31, K=2,3) -> inner loop is one
// ds_load_b64 + wmma + min-accumulate, no per-iteration selects.
// ---------------------------------------------------------------------------
__global__ void __launch_bounds__(256) chamfer_colmin(
    const float* __restrict__ pts, float* __restrict__ out,
    int n_pts, int m_pix) {
    __shared__ v2f sfrag[2][2048];  // 32 KB
    for (int i = threadIdx.x; i < n_pts; i += blockDim.x) {
        float x = pts[2 * i], y = pts[2 * i + 1];
        v2f xy; xy.x = x; xy.y = y;
        v2f q;  q.x = x * x + y * y; q.y = 1.0f;
        sfrag[0][i] = xy;
        sfrag[1][i] = q;
    }
    __syncthreads();

    const int wave = (blockIdx.x * blockDim.x + threadIdx.x) >> 5;  // pixel tile
    const int lane = threadIdx.x & 31;
    const int l    = lane & 15;
    const int half = lane >> 4;

    // this wave's 16 pixels: linear index m, x = m % 256, y = m / 256
    const int   m  = wave * 16 + l;
    const float cx = (float)(m & (W_IMG - 1));
    const float cy = (float)(m >> 8);
    const float cc = cx * cx + cy * cy;

    // B 4x16 layout: lane<16 -> K=v, lane>=16 -> K=v+2
    v2f b;
    b.x = half ? 1.0f : -2.0f * cx;   // K=0 / K=2
    b.y = half ? cc   : -2.0f * cy;   // K=1 / K=3

    v8f acc;
#pragma unroll
    for (int v = 0; v < 8; ++v) acc[v] = BIG;
    v8f c0 = {};

    const v2f* __restrict__ lanebase = &sfrag[half][l];
    const int ntiles = n_pts >> 4;  // 125
    for (int t = 0; t < ntiles; ++t) {
        v2f a = lanebase[t * 16];   // single ds_load_b64
        v8f d = __builtin_amdgcn_wmma_f32_16x16x4_f32(
            false, a, false, b, (short)0, c0, false, false);
#pragma unroll
        for (int v = 0; v < 8; ++v) acc[v] = fminf(acc[v], d[v]);
    }

    // column (per-pixel) min: 8 rows in-lane, then rows 8..15 via xor-16
    float m8 = acc[0];
#pragma unroll
    for (int v = 1; v < 8; ++v) m8 = fminf(m8, acc[v]);
    m8 = fminf(m8, __shfl_xor(m8, 16, 32));

    float s = sqrtf(fmaxf(m8, 0.0f));
#pragma unroll
    for (int off = 1; off < 16; off <<= 1) s += __shfl_xor(s, off, 32);
    if (lane == 0) atomicAdd(out, s);  // one atomic per wave
}

// ---------------------------------------------------------------------------
// Kernel B: per-point min over all pixels. One block = one 16-point tile
// (A fragment fixed), its 8 waves stripe the 4096 pixel tiles by 8.
// Grid structure: for stride-8 tile walk with W=256, cx alternates between
// two per-lane constants (cxA, cxB) while cy increments every 2 tiles, so
// iterate cy=0..15 with 2 WMMAs per step and O(1) fragment update.
// ---------------------------------------------------------------------------
__global__ void __launch_bounds__(256) chamfer_rowmin(
    const float* __restrict__ pts, unsigned int* __restrict__ rowmin,
    int n_pts, int m_pix) {
    const int lane  = threadIdx.x & 31;
    const int chunk = threadIdx.x >> 5;  // 0..7
    const int l     = lane & 15;
    const int half  = lane >> 4;
    const int tile  = blockIdx.x;        // point tile

    const int pi = tile * 16 + l;
    const float px = pts[2 * pi], py = pts[2 * pi + 1];
    const float pp = px * px + py * py;
    v2f a;
    a.x = half ? pp   : px;
    a.y = half ? 1.0f : py;

    // tiles visited: p = chunk + 8j.  p%16 in {chunk, chunk+8}; cy = p/16.
    const float cxA  = (float)(chunk * 16 + l);
    const float cxB  = (float)((chunk + 8) * 16 + l);
    const float cxA2 = cxA * cxA;
    const float cxB2 = cxB * cxB;
    const float bxA  = half ? 1.0f : -2.0f * cxA;  // loop-invariant K=0/K=2
    const float bxB  = half ? 1.0f : -2.0f * cxB;

    v8f accA, accB;
#pragma unroll
    for (int v = 0; v < 8; ++v) { accA[v] = BIG; accB[v] = BIG; }
    v8f c0 = {};

    const int nrows = m_pix >> 12;  // (m_pix/16)/256 = 16 rows of tiles
    for (int k = 0; k < nrows; ++k) {
        const float cyf = (float)k;
        const float cy2 = cyf * cyf;
        v2f bA, bB;
        bA.x = bxA;
        bB.x = bxB;
        bA.y = half ? (cxA2 + cy2) : -2.0f * cyf;
        bB.y = half ? (cxB2 + cy2) : -2.0f * cyf;
        v8f dA = __builtin_amdgcn_wmma_f32_16x16x4_f32(
            false, a, false, bA, (short)0, c0, false, false);
        v8f dB = __builtin_amdgcn_wmma_f32_16x16x4_f32(
            false, a, false, bB, (short)0, c0, false, false);
#pragma unroll
        for (int v = 0; v < 8; ++v) {
            accA[v] = fminf(accA[v], dA[v]);
            accB[v] = fminf(accB[v], dB[v]);
        }
    }
#pragma unroll
    for (int v = 0; v < 8; ++v) accA[v] = fminf(accA[v], accB[v]);

    // row (per-point) min: reduce over the 16 columns (lanes in 16-group)
#pragma unroll
    for (int v = 0; v < 8; ++v) {
        float x = accA[v];
#pragma unroll
        for (int off = 1; off < 16; off <<= 1)
            x = fminf(x, __shfl_xor(x, off, 32));
        accA[v] = x;
    }
    if (l == 0) {
#pragma unroll
        for (int v = 0; v < 8; ++v) {
            const int   row = tile * 16 + v + 8 * half;
            const float val = fmaxf(accA[v], 0.0f);  // clamp => uint order valid
            atomicMin(&rowmin[row], __float_as_uint(val));
        }
    }
}

// ---------------------------------------------------------------------------
// final: out += sum_i sqrt(rowmin[i])
// ---------------------------------------------------------------------------
__global__ void __launch_bounds__(256) chamfer_rowsum(
    const unsigned int* __restrict__ rowmin, float* out, int n) {
    __shared__ float red[256];
    float s = 0.0f;
    for (int i = threadIdx.x; i < n; i += 256)
        s += sqrtf(fmaxf(__uint_as_float(rowmin[i]), 0.0f));
    red[threadIdx.x] = s;
    __syncthreads();
    for (int step = 128; step > 0; step >>= 1) {
        if (threadIdx.x < step) red[threadIdx.x] += red[threadIdx.x + step];
        __syncthreads();
    }
    if (threadIdx.x == 0) atomicAdd(out, red[0]);
}

extern "C" void kernel_launch(void* const* d_in, const int* in_sizes, int n_in,
                              void* d_out, int out_size, void* d_ws, size_t ws_size,
                              hipStream_t stream) {
    const float* pts = (const float*)d_in[0];   // (N,2) float32
    const int n_pts = in_sizes[0] / 2;          // 2000
    const int m_pix = in_sizes[1];              // 65536 (H*W of img_ref)
    float* out = (float*)d_out;
    unsigned int* rowmin = (unsigned int*)d_ws; // 8 KB

    chamfer_init<<<(n_pts + 255) / 256, 256, 0, stream>>>(out, rowmin, n_pts);

    const int pix_waves = m_pix / 16;           // 4096 pixel tiles
    chamfer_colmin<<<pix_waves / 8, 256, 0, stream>>>(pts, out, n_pts, m_pix);
    chamfer_rowmin<<<n_pts / 16, 256, 0, stream>>>(pts, rowmin, n_pts, m_pix);
    chamfer_rowsum<<<1, 256, 0, stream>>>(rowmin, out, n_pts);
}